// HexGraphConv_22488448762244
// MI455X (gfx1250) — compile-verified
//
#include <hip/hip_runtime.h>

typedef __attribute__((ext_vector_type(2))) float v2f;
typedef __attribute__((ext_vector_type(8))) float v8f;

#define D 128
#define NEG_SLOPE 0.1f

// Guaranteed-hardware fp32 atomic add (CDNA5 GLOBAL_ATOMIC_ADD_F32,
// non-returning => TH=0, device scope so L2 atomic units are the
// coherence point visible to the following kernel launch).
__device__ __forceinline__ void hw_atomic_add_f32(float* p, float v) {
#if defined(__gfx1250__)
    asm volatile("global_atomic_add_f32 %0, %1, off scope:SCOPE_DEV"
                 :: "v"(p), "v"(v) : "memory");
#else
    atomicAdd(p, v);
#endif
}

// ---------------------------------------------------------------------------
// Stage 1: scatter-add raw neighbor features (pre-linear!) + edge counts.
// 32 lanes per edge, 4 floats per lane (float4 gather, 4 fp32 atomics).
// ---------------------------------------------------------------------------
__global__ __launch_bounds__(256) void hex_scatter(
    const float* __restrict__ x,          // [N, 128]
    const long long* __restrict__ ei,     // [2, E] int64
    float* __restrict__ agg,              // [N, 128] accumulator (ws)
    float* __restrict__ cnt,              // [N] edge counts (ws)
    int E)
{
    long long t = (long long)blockIdx.x * blockDim.x + threadIdx.x;
    int e = (int)(t >> 5);
    if (e >= E) return;
    int sub = (int)(t & 31);

    long long s = ei[e];                  // src row
    long long d = ei[(long long)E + e];   // dst row

    const float4 v = *(const float4*)(x + s * D + sub * 4);
    float* base = agg + d * D + sub * 4;
    hw_atomic_add_f32(base + 0, v.x);
    hw_atomic_add_f32(base + 1, v.y);
    hw_atomic_add_f32(base + 2, v.z);
    hw_atomic_add_f32(base + 3, v.w);
    if (sub == 0) hw_atomic_add_f32(cnt + d, 1.0f);
}

// ---------------------------------------------------------------------------
// Stage 2: fused dual-GEMM + bias + degree-normalize + LeakyReLU via
// V_WMMA_F32_16X16X4_F32. One wave handles a 16-node row tile.
//   out = LeakyReLU( x@Ws^T + (S/denom)@Wn^T + self_b + (cnt/denom)*neigh_b )
// Both GEMMs chain into the SAME accumulator (row-scaling of S commutes
// with the GEMM and is applied while staging into LDS).
// ---------------------------------------------------------------------------
__global__ __launch_bounds__(32) void hex_fused_gemm(
    const float* __restrict__ x,          // [N, 128]
    const float* __restrict__ aggS,       // [N, 128] scatter result
    const float* __restrict__ cnt,        // [N]
    const long long* __restrict__ deg,    // [N] int64
    const float* __restrict__ Ws,         // self_W [128,128] row-major [o][d]
    const float* __restrict__ sb,         // self_b [128]
    const float* __restrict__ Wn,         // neigh_W [128,128]
    const float* __restrict__ nb,         // neigh_b [128]
    float* __restrict__ out,              // [N, 128]
    int N)
{
    __shared__ float lx[16 * D];          // x tile
    __shared__ float ls[16 * D];          // (agg/denom) tile
    __shared__ float linv[16];            // 1/denom per row
    __shared__ float lcs[16];             // cnt/denom per row

    const int lane  = threadIdx.x;
    const int node0 = blockIdx.x * 16;
    if (node0 >= N) return;

    if (lane < 16) {
        float dg = (float)deg[node0 + lane];
        float denom = dg > 1.0f ? dg : 1.0f;
        float inv = 1.0f / denom;
        linv[lane] = inv;
        lcs[lane]  = cnt[node0 + lane] * inv;
    }
    __syncthreads();

    // Stage A tiles into LDS (float4 per lane-iteration; 16 iters).
    for (int i = lane; i < 16 * 32; i += 32) {
        int row = i >> 5;
        int c4  = (i & 31) << 2;
        size_t goff = (size_t)(node0 + row) * D + c4;
        *(float4*)(lx + row * D + c4) = *(const float4*)(x + goff);
        float4 vs = *(const float4*)(aggS + goff);
        float inv = linv[row];
        vs.x *= inv; vs.y *= inv; vs.z *= inv; vs.w *= inv;
        *(float4*)(ls + row * D + c4) = vs;
    }
    __syncthreads();

    // Per-lane WMMA addressing constants (f32 16x16x4 layout):
    //   A: lanes 0-15 hold K=0,1 ; lanes 16-31 hold K=2,3 (row = lane&15)
    //   B: mirrored with N = lane&15
    //   C/D: VGPR r -> M = r + (lane<16 ? 0 : 8), N = lane&15
    const int rsel = lane & 15;
    const int koff = (lane >> 4) << 1;    // 0 or 2
    const int mbas = (lane >> 4) << 3;    // 0 or 8

    float csv[8];
    #pragma unroll
    for (int r = 0; r < 8; ++r) csv[r] = lcs[mbas + r];

    const float* lxr = lx + rsel * D + koff;
    const float* lsr = ls + rsel * D + koff;

    #pragma unroll 1
    for (int ot = 0; ot < 8; ++ot) {
        const int o = ot * 16 + rsel;
        const float sbv = sb[o];
        const float nbv = nb[o];

        v8f c;
        #pragma unroll
        for (int r = 0; r < 8; ++r) c[r] = sbv + csv[r] * nbv;

        const float* wsp = Ws + (size_t)o * D + koff;
        const float* wnp = Wn + (size_t)o * D + koff;

        // GEMM 1: x @ self_W^T
        #pragma unroll
        for (int kk = 0; kk < 32; ++kk) {
            v2f a; a.x = lxr[kk * 4];     a.y = lxr[kk * 4 + 1];
            v2f b; b.x = wsp[kk * 4];     b.y = wsp[kk * 4 + 1];
            c = __builtin_amdgcn_wmma_f32_16x16x4_f32(
                    false, a, false, b, (short)0, c, false, false);
        }
        // GEMM 2: (agg/denom) @ neigh_W^T, chained into same accumulator
        #pragma unroll
        for (int kk = 0; kk < 32; ++kk) {
            v2f a; a.x = lsr[kk * 4];     a.y = lsr[kk * 4 + 1];
            v2f b; b.x = wnp[kk * 4];     b.y = wnp[kk * 4 + 1];
            c = __builtin_amdgcn_wmma_f32_16x16x4_f32(
                    false, a, false, b, (short)0, c, false, false);
        }

        // LeakyReLU + store
        #pragma unroll
        for (int r = 0; r < 8; ++r) {
            float y = c[r];
            y = y > 0.0f ? y : NEG_SLOPE * y;
            out[(size_t)(node0 + mbas + r) * D + o] = y;
        }
    }
}

// ---------------------------------------------------------------------------
extern "C" void kernel_launch(void* const* d_in, const int* in_sizes, int n_in,
                              void* d_out, int out_size, void* d_ws, size_t ws_size,
                              hipStream_t stream) {
    const float*     x   = (const float*)d_in[0];
    const long long* ei  = (const long long*)d_in[1];
    const long long* deg = (const long long*)d_in[2];
    const float*     Ws  = (const float*)d_in[3];
    const float*     sb  = (const float*)d_in[4];
    const float*     Wn  = (const float*)d_in[5];
    const float*     nb  = (const float*)d_in[6];
    float* out = (float*)d_out;

    const int E = in_sizes[1] / 2;
    const int N = in_sizes[2];

    float* agg = (float*)d_ws;            // [N, 128]
    float* cnt = agg + (size_t)N * D;     // [N]
    size_t zero_bytes = ((size_t)N * D + N) * sizeof(float);

    hipMemsetAsync(d_ws, 0, zero_bytes, stream);

    {   // scatter: 32 lanes per edge
        long long threads = (long long)E * 32;
        int blk = 256;
        int grid = (int)((threads + blk - 1) / blk);
        hex_scatter<<<grid, blk, 0, stream>>>(x, ei, agg, cnt, E);
    }
    {   // fused GEMM: 16 nodes per wave
        int grid = (N + 15) / 16;
        hex_fused_gemm<<<grid, 32, 0, stream>>>(x, agg, cnt, deg,
                                                Ws, sb, Wn, nb, out, N);
    }
}